// MONForwardBackwardSplitting_70300024701178
// MI455X (gfx1250) — compile-verified
//
#include <hip/hip_runtime.h>

typedef __attribute__((ext_vector_type(2))) float v2f;
typedef __attribute__((ext_vector_type(8))) float v8f;

#define BSZ   32
#define CCH   128
#define HH    32
#define WW    32
#define LL    (CCH * HH * WW)   // 131072 = 2^17
#define KK    (CCH * 9)         // 1152
#define KCH   576               // K per chunk (64 input channels * 9)
#define PADK  580               // padded LDS row stride (580 % 64 == 4 -> conflict-free)
#define ALPHA_ 0.9f
#define LAM_   1e-4f
#define TOL_   1e-5f

// ---------------- elementwise helpers ----------------

__global__ void zero_f(float* p, long n) {
  long i = (long)blockIdx.x * blockDim.x + threadIdx.x;
  long st = (long)gridDim.x * blockDim.x;
  for (; i < n; i += st) p[i] = 0.f;
}

__global__ void init_misc(int* done) {
  if (threadIdx.x == 0 && blockIdx.x == 0) *done = 0;
}

__global__ void fill_slice(float* base, long bstride, float v) {
  long i = (long)blockIdx.x * blockDim.x + threadIdx.x;
  if (i >= (long)BSZ * LL) return;
  long b = i >> 17, l = i & (LL - 1);
  base[b * bstride + l] = v;
}

__global__ void copy_slice(float* dst, long dstride, const float* src, long sstride) {
  long i = (long)blockIdx.x * blockDim.x + threadIdx.x;
  if (i >= (long)BSZ * LL) return;
  long b = i >> 17, l = i & (LL - 1);
  dst[b * dstride + l] = src[b * sstride + l];
}

// W[co][k] (co-major, k = ci*9+kh*3+kw) -> Wtp[k/2][co][2]  (pair-interleaved so
// {W[k][co], W[k+1][co]} is one aligned 8-byte load)
__global__ void wt_pack_pairs(const float* __restrict__ W, float* __restrict__ Wtp) {
  int i = blockIdx.x * 256 + threadIdx.x;
  if (i >= KK * CCH) return;
  int k = i >> 7, co = i & 127;
  int dst = ((k >> 1) << 8) + (co << 1) + (k & 1);   // (k/2)*256 + co*2 + k%2
  Wtp[dst] = W[(long)co * KK + k];
}

// zm = z*(1-mask) + mask*x   (mask is [B,1,H,W] int, broadcast over C)
__global__ void zm_kernel(const float* __restrict__ z, long zbs,
                          const float* __restrict__ x, const int* __restrict__ mask,
                          float* __restrict__ zm, const int* done, int gate) {
  if (gate && *done) return;
  long i = (long)blockIdx.x * blockDim.x + threadIdx.x;
  if (i >= (long)BSZ * LL) return;
  long b = i >> 17, rem = i & (LL - 1);
  long sp = rem & (HH * WW - 1);
  float mf = (float)mask[b * (HH * WW) + sp];
  zm[i] = z[b * zbs + rem] * (1.f - mf) + mf * x[i];
}

// ---------------- WMMA conv + damping + channel softmax ----------------
// Block = 256 threads = 8 waves. Block handles one (n, h, w0..w0+15) strip,
// wave wv handles output channels [wv*16, wv*16+16).
// GEMM per wave: D[p(16) x co(16)] += A[p x 4] * B[4 x co] over K=1152,
// processed in two K-chunks of 576 staged as im2col [p][k] in LDS.
__global__ __launch_bounds__(256)
void conv_softmax(const float* __restrict__ zm, const float* __restrict__ Wtp,
                  const float* __restrict__ bias,
                  const float* __restrict__ z, long zbs,
                  float* __restrict__ out, long obs,
                  const int* done, int gate) {
  if (gate && *done) return;

  __shared__ __align__(16) float patchK[16 * PADK];  // im2col chunk: [p][k], padded
  __shared__ __align__(16) float pre[16 * CCH];      // [p][co]
  __shared__ float red[256];
  __shared__ float pmax[16];
  __shared__ float psum[16];

  const int blk = blockIdx.x;
  const int n   = blk >> 6;
  const int rem = blk & 63;
  const int h   = rem >> 1;
  const int w0  = (rem & 1) << 4;
  const int tid = threadIdx.x;

  const int lane    = tid & 31;
  const int wv      = tid >> 5;
  const int co_base = wv << 4;
  const int p       = lane & 15;        // A: M = position; B/D: N index
  const int khalf   = lane >> 4;        // K sub-pair select (0 -> k0..k0+1, 1 -> k0+2..k0+3)
  const int co_n    = co_base + p;      // B-matrix column (channel)

  const float* aRow = patchK + (size_t)p * PADK;   // per-lane A row base

  v8f acc = {0.f, 0.f, 0.f, 0.f, 0.f, 0.f, 0.f, 0.f};

  for (int ch = 0; ch < 2; ++ch) {
    __syncthreads();   // all waves done reading previous chunk
    // Build im2col chunk: patchK[pp][kk] = zm[n, ci, h+kh-1, w0+pp+kw-1] (zero halo)
    for (int idx = tid; idx < 16 * KCH; idx += 256) {
      int kk = idx >> 4;            // 0..575
      int pp = idx & 15;
      int k  = ch * KCH + kk;
      int ci = k / 9;
      int r  = k - ci * 9;
      int kh = r / 3, kw = r - kh * 3;
      int hh = h - 1 + kh, ww = w0 - 1 + kw + pp;
      float v = 0.f;
      if (hh >= 0 && hh < HH && ww >= 0 && ww < WW)
        v = zm[(((long)n * CCH + ci) * HH + hh) * WW + ww];
      patchK[pp * PADK + kk] = v;
    }
    __syncthreads();

    const float* wBase = Wtp + (size_t)(ch * KCH) * CCH;  // pair-interleaved chunk base
    for (int kk = 0; kk < KCH; kk += 4) {
      int k = kk + (khalf << 1);     // even
      v2f a  = *(const v2f*)(aRow + k);                               // ds_load_b64
      v2f bf = *(const v2f*)(wBase + ((size_t)(k >> 1) << 8) + (co_n << 1)); // global b64
      acc = __builtin_amdgcn_wmma_f32_16x16x4_f32(false, a, false, bf,
                                                  (short)0, acc, false, false);
    }
  }

  // pre = (1-alpha)*z + alpha*(lin + bias); D layout: VGPR v -> M=v+8*khalf, N=lane&15
  const float oma = 1.f - ALPHA_;
#pragma unroll
  for (int v = 0; v < 8; ++v) {
    int prow = v + (khalf << 3);
    int coc  = co_base + (lane & 15);
    float lin  = acc[v] + bias[coc];
    float zval = z[(long)n * zbs + ((long)coc * HH + h) * WW + w0 + prow];
    pre[prow * CCH + coc] = oma * zval + ALPHA_ * lin;
  }
  __syncthreads();

  // softmax over 128 channels per position; thread t owns (p = t&15, 8 channels)
  const int sp = tid & 15;
  const int cg = tid >> 4;
  const int c0 = cg << 3;
  float lm = -1e30f;
  for (int c = c0; c < c0 + 8; ++c) lm = fmaxf(lm, pre[sp * CCH + c]);
  red[cg * 16 + sp] = lm;
  __syncthreads();
  if (tid < 16) {
    float m = -1e30f;
    for (int g = 0; g < 16; ++g) m = fmaxf(m, red[g * 16 + tid]);
    pmax[tid] = m;
  }
  __syncthreads();
  const float mm = pmax[sp];
  float ls = 0.f;
  for (int c = c0; c < c0 + 8; ++c) {
    float e = expf(pre[sp * CCH + c] - mm);
    pre[sp * CCH + c] = e;
    ls += e;
  }
  __syncthreads();
  red[cg * 16 + sp] = ls;
  __syncthreads();
  if (tid < 16) {
    float s = 0.f;
    for (int g = 0; g < 16; ++g) s += red[g * 16 + tid];
    psum[tid] = s;
  }
  __syncthreads();
  const float inv = 1.f / psum[sp];
  for (int c = c0; c < c0 + 8; ++c)
    out[(long)n * obs + ((long)c * HH + h) * WW + w0 + sp] = pre[sp * CCH + c] * inv;
}

// ---------------- Anderson machinery ----------------

// Gram[b][i][j] = sum_l (Fh[b,i]-X[b,i])(Fh[b,j]-X[b,j]); one block per b, fixed-order reduce.
__global__ void gram_kernel(const float* __restrict__ X, const float* __restrict__ Fh,
                            float* __restrict__ gram, const int* done) {
  if (*done) return;
  __shared__ float red[256];
  int b = blockIdx.x, tid = threadIdx.x;
  float acc[15];
#pragma unroll
  for (int s = 0; s < 15; ++s) acc[s] = 0.f;
  for (long l = tid; l < LL; l += 256) {
    float g[5];
#pragma unroll
    for (int i = 0; i < 5; ++i)
      g[i] = Fh[((long)b * 5 + i) * LL + l] - X[((long)b * 5 + i) * LL + l];
    int idx = 0;
#pragma unroll
    for (int i = 0; i < 5; ++i)
#pragma unroll
      for (int j = i; j < 5; ++j) acc[idx++] += g[i] * g[j];
  }
  int idx = 0;
  for (int i = 0; i < 5; ++i)
    for (int j = i; j < 5; ++j) {
      red[tid] = acc[idx];
      __syncthreads();
      for (int s = 128; s > 0; s >>= 1) {
        if (tid < s) red[tid] += red[tid + s];
        __syncthreads();
      }
      if (tid == 0) {
        gram[b * 25 + i * 5 + j] = red[0];
        gram[b * 25 + j * 5 + i] = red[0];
      }
      __syncthreads();
      ++idx;
    }
}

// Bordered 6x6 solve, partial pivoting (H[0][0]=0 requires it). One thread per batch.
__global__ void solve_kernel(const float* __restrict__ gram, float* __restrict__ alpha,
                             int nact, const int* done) {
  if (*done) return;
  int b = threadIdx.x;
  if (b >= BSZ) return;
  float Hm[6][6], y[6];
  for (int i = 0; i < 6; ++i) {
    y[i] = (i == 0) ? 1.f : 0.f;
    for (int j = 0; j < 6; ++j) Hm[i][j] = 0.f;
  }
  for (int i = 0; i < 5; ++i) {
    float ai = (i < nact) ? 1.f : 0.f;
    Hm[0][i + 1] = ai;
    Hm[i + 1][0] = ai;
    for (int j = 0; j < 5; ++j) {
      float hc = gram[b * 25 + i * 5 + j] + ((i == j) ? LAM_ : 0.f);
      float m2 = ((i < nact) && (j < nact)) ? 1.f : 0.f;
      hc = hc * m2 + (((i == j) && (i >= nact)) ? 1.f : 0.f);
      Hm[i + 1][j + 1] = hc;
    }
  }
  for (int col = 0; col < 6; ++col) {
    int piv = col;
    float mx = fabsf(Hm[col][col]);
    for (int r = col + 1; r < 6; ++r) {
      float v = fabsf(Hm[r][col]);
      if (v > mx) { mx = v; piv = r; }
    }
    if (piv != col) {
      for (int c = 0; c < 6; ++c) {
        float t = Hm[col][c]; Hm[col][c] = Hm[piv][c]; Hm[piv][c] = t;
      }
      float t = y[col]; y[col] = y[piv]; y[piv] = t;
    }
    float invd = 1.f / Hm[col][col];
    for (int r = col + 1; r < 6; ++r) {
      float f = Hm[r][col] * invd;
      for (int c = col; c < 6; ++c) Hm[r][c] -= f * Hm[col][c];
      y[r] -= f * y[col];
    }
  }
  for (int col = 5; col >= 0; --col) {
    float s = y[col];
    for (int c = col + 1; c < 6; ++c) s -= Hm[col][c] * y[c];
    y[col] = s / Hm[col][col];
  }
  for (int j = 0; j < 5; ++j) alpha[b * 8 + j] = y[1 + j];
}

// xnew = sum_j alpha_j * Fh[b,j]  (BETA == 1.0); writes X[:,slot] and zstar
__global__ void xnew_kernel(const float* __restrict__ Fh, const float* __restrict__ alpha,
                            float* __restrict__ Xslot, long xbs,
                            float* __restrict__ zstar, const int* done) {
  if (*done) return;
  long i = (long)blockIdx.x * blockDim.x + threadIdx.x;
  if (i >= (long)BSZ * LL) return;
  long b = i >> 17, l = i & (LL - 1);
  float s = 0.f;
#pragma unroll
  for (int j = 0; j < 5; ++j)
    s += alpha[b * 8 + j] * Fh[((long)b * 5 + j) * LL + l];
  Xslot[b * xbs + l] = s;
  zstar[b * (long)LL + l] = s;
}

__global__ void resid_partial(const float* __restrict__ fnew, const float* __restrict__ xnew,
                              long bstride, float* __restrict__ partials, const int* done) {
  if (*done) return;
  __shared__ float r1[256], r2[256];
  int blk = blockIdx.x, tid = threadIdx.x;
  float s1 = 0.f, s2 = 0.f;
  long base = (long)blk * 4096;
  for (int it = 0; it < 16; ++it) {
    long g = base + it * 256 + tid;
    long b = g >> 17, l = g & (LL - 1);
    float fn = fnew[b * bstride + l];
    float xn = xnew[b * bstride + l];
    float d = fn - xn;
    s1 += d * d;
    s2 += fn * fn;
  }
  r1[tid] = s1; r2[tid] = s2;
  __syncthreads();
  for (int s = 128; s > 0; s >>= 1) {
    if (tid < s) { r1[tid] += r1[tid + s]; r2[tid] += r2[tid + s]; }
    __syncthreads();
  }
  if (tid == 0) { partials[blk] = r1[0]; partials[1024 + blk] = r2[0]; }
}

__global__ void resid_final(const float* __restrict__ partials, int* done) {
  if (*done) return;
  __shared__ float r1[256], r2[256];
  int tid = threadIdx.x;
  float s1 = 0.f, s2 = 0.f;
  for (int i = tid; i < 1024; i += 256) { s1 += partials[i]; s2 += partials[1024 + i]; }
  r1[tid] = s1; r2[tid] = s2;
  __syncthreads();
  for (int s = 128; s > 0; s >>= 1) {
    if (tid < s) { r1[tid] += r1[tid + s]; r2[tid] += r2[tid + s]; }
    __syncthreads();
  }
  if (tid == 0) {
    float res = sqrtf(r1[0]) / (1e-5f + sqrtf(r2[0]));
    if (res < TOL_) *done = 1;
  }
}

// ---------------- host orchestration ----------------

extern "C" void kernel_launch(void* const* d_in, const int* in_sizes, int n_in,
                              void* d_out, int out_size, void* d_ws, size_t ws_size,
                              hipStream_t stream) {
  const float* x    = (const float*)d_in[0];
  const float* W    = (const float*)d_in[1];
  const float* bias = (const float*)d_in[2];
  const int*   mask = (const int*)d_in[3];
  float* out = (float*)d_out;

  const long L = LL;
  float* ws = (float*)d_ws;
  size_t off = 0;
  float* Xh    = ws + off; off += (size_t)BSZ * 5 * L;
  float* Fh    = ws + off; off += (size_t)BSZ * 5 * L;
  float* zm    = ws + off; off += (size_t)BSZ * L;
  float* zstar = ws + off; off += (size_t)BSZ * L;
  float* Wtp   = ws + off; off += (size_t)KK * CCH;
  float* gram  = ws + off; off += BSZ * 25;
  float* alpha = ws + off; off += BSZ * 8;
  float* parts = ws + off; off += 2048;
  int*   done  = (int*)(ws + off);

  dim3 b256(256);
  const long totE = (long)BSZ * L;                 // 4,194,304
  const int gridE = (int)((totE + 255) / 256);     // 16384
  const long hist = (long)BSZ * 5 * L;

  wt_pack_pairs<<<(KK * CCH + 255) / 256, b256, 0, stream>>>(W, Wtp);
  zero_f<<<4096, b256, 0, stream>>>(Xh, hist);
  zero_f<<<4096, b256, 0, stream>>>(Fh, hist);
  init_misc<<<1, 1, 0, stream>>>(done);
  fill_slice<<<gridE, b256, 0, stream>>>(Xh, 5 * L, 1.0f / CCH);   // x0 = uniform 1/C

  // F0 = f(x0): Fh[:,0]
  zm_kernel<<<gridE, b256, 0, stream>>>(Xh, 5 * L, x, mask, zm, done, 0);
  conv_softmax<<<2048, b256, 0, stream>>>(zm, Wtp, bias, Xh, 5 * L, Fh, 5 * L, done, 0);
  // X[:,1] = F0
  copy_slice<<<gridE, b256, 0, stream>>>(Xh + L, 5 * L, Fh, 5 * L);
  // F1 = f(F0): Fh[:,1]
  zm_kernel<<<gridE, b256, 0, stream>>>(Fh, 5 * L, x, mask, zm, done, 0);
  conv_softmax<<<2048, b256, 0, stream>>>(zm, Wtp, bias, Fh, 5 * L, Fh + L, 5 * L, done, 0);

  for (int k = 2; k < 50; ++k) {
    int nact = (k < 5) ? k : 5;
    long slot = k % 5;
    gram_kernel<<<BSZ, b256, 0, stream>>>(Xh, Fh, gram, done);
    solve_kernel<<<1, 32, 0, stream>>>(gram, alpha, nact, done);
    xnew_kernel<<<gridE, b256, 0, stream>>>(Fh, alpha, Xh + slot * L, 5 * L, zstar, done);
    zm_kernel<<<gridE, b256, 0, stream>>>(Xh + slot * L, 5 * L, x, mask, zm, done, 1);
    conv_softmax<<<2048, b256, 0, stream>>>(zm, Wtp, bias, Xh + slot * L, 5 * L,
                                            Fh + slot * L, 5 * L, done, 1);
    resid_partial<<<1024, b256, 0, stream>>>(Fh + slot * L, Xh + slot * L, 5 * L, parts, done);
    resid_final<<<1, b256, 0, stream>>>(parts, done);
  }

  // zn = f(zstar) -> d_out (always runs)
  zm_kernel<<<gridE, b256, 0, stream>>>(zstar, L, x, mask, zm, done, 0);
  conv_softmax<<<2048, b256, 0, stream>>>(zm, Wtp, bias, zstar, L, out, L, done, 0);
}